// GCNN_23948737642596
// MI455X (gfx1250) — compile-verified
//
#include <hip/hip_runtime.h>
#include <math.h>

// GCN 2-layer forward for MI455X (gfx1250, wave32).
// Working set (~130MB) fits in the 192MB L2 -> scatter atomics dominate, GEMM is
// trivial; we still run h1 = x@W1 on the fp32 WMMA path (V_WMMA_F32_16X16X4_F32)
// which is bit-compatible with the fp32 reference.

#define N_NODES 100000
#define N_EDGES 3200000
#define D_IN    128
#define D_HID   64
#define D_OUT   4

typedef float v2f __attribute__((ext_vector_type(2)));
typedef float v8f __attribute__((ext_vector_type(8)));

// ---------------- degree / normalization ----------------
__global__ __launch_bounds__(256) void k_init_deg(float* __restrict__ deg) {
    int i = blockIdx.x * 256 + threadIdx.x;
    if (i < N_NODES) deg[i] = 1.0f;               // self-loop weight
}

__global__ __launch_bounds__(256) void k_edge_deg(const int* __restrict__ dst,
                                                  const float* __restrict__ w,
                                                  float* __restrict__ deg) {
    int e = blockIdx.x * 256 + threadIdx.x;
    if (e < N_EDGES)
        __hip_atomic_fetch_add(&deg[dst[e]], w[e], __ATOMIC_RELAXED,
                               __HIP_MEMORY_SCOPE_AGENT);
}

__global__ __launch_bounds__(256) void k_dinv(float* __restrict__ deg) {
    int i = blockIdx.x * 256 + threadIdx.x;
    if (i < N_NODES) {
        float d = deg[i];
        deg[i] = (d > 0.0f) ? rsqrtf(d) : 0.0f;   // in place: deg -> dinv
    }
}

// ---------------- layer-1 GEMM: h1 = x @ W1 (fp32 WMMA) ----------------
// Block = 256 threads = 8 waves; block computes a 32x64 tile of h1.
// Wave w: rows [(w>>2)*16, +16), cols [(w&3)*16, +16); K-loop in steps of 4
// using V_WMMA_F32_16X16X4_F32.  x tile and W1 staged in LDS with padded
// strides to kill bank conflicts (sX stride 130, sW stride 72).
#define SX_STRIDE 130
#define SW_STRIDE 72
__global__ __launch_bounds__(256) void k_gemm1(const float* __restrict__ x,
                                               const float* __restrict__ W1,
                                               float* __restrict__ h1) {
    __shared__ float sW[D_IN * SW_STRIDE];   // 128x64 padded to stride 72 (36 KB)
    __shared__ float sX[32 * SX_STRIDE];     // 32x128 padded to stride 130 (~16.6 KB)

    const int tid = threadIdx.x;
    const int R0  = blockIdx.x * 32;         // 100000 = 32 * 3125 exactly

    for (int i = tid; i < D_IN * D_HID; i += 256)
        sW[(i >> 6) * SW_STRIDE + (i & 63)] = W1[i];
    for (int i = tid; i < 32 * D_IN; i += 256)
        sX[(i >> 7) * SX_STRIDE + (i & 127)] = x[R0 * D_IN + i];
    __syncthreads();

    const int lane = tid & 31;
    const int wv   = tid >> 5;
    const int lo   = lane & 15;
    const int hi   = lane >> 4;
    const int mloc = (wv >> 2) << 4;         // 0 or 16
    const int col0 = (wv & 3) << 4;          // 0,16,32,48

    v8f c = {};
    const float* xrow = &sX[(mloc + lo) * SX_STRIDE];
#pragma unroll
    for (int k = 0; k < D_IN; k += 4) {
        const int ka = k + (hi << 1);        // lanes 0-15: K=k,k+1; 16-31: K=k+2,k+3
        v2f a = *(const v2f*)&xrow[ka];
        v2f b;
        b.x = sW[ka * SW_STRIDE + col0 + lo];
        b.y = sW[ka * SW_STRIDE + SW_STRIDE + col0 + lo];
        c = __builtin_amdgcn_wmma_f32_16x16x4_f32(
                /*neg_a=*/false, a, /*neg_b=*/false, b,
                /*c_mod=*/(short)0, c, /*reuse_a=*/false, /*reuse_b=*/false);
    }

    const int gr = R0 + mloc + (hi << 3);    // C VGPR j holds M = j + 8*hi
#pragma unroll
    for (int j = 0; j < 8; ++j)
        h1[(gr + j) * D_HID + col0 + lo] = c[j];
}

// ---------------- layer-1 scatter: agg1[dst] += norm * h1[src] ----------------
// 16 lanes per edge, each handles 4 contiguous dims via a float4 gather.
__global__ __launch_bounds__(256) void k_scatter1(const int* __restrict__ src,
                                                  const int* __restrict__ dst,
                                                  const float* __restrict__ w,
                                                  const float* __restrict__ dinv,
                                                  const float* __restrict__ h1,
                                                  float* __restrict__ agg1) {
    int g = blockIdx.x * 256 + threadIdx.x;  // E*16 = 51.2M threads
    int e = g >> 4;
    if (e < N_EDGES) {
        int q = (g & 15) << 2;
        int s = src[e], t = dst[e];
        float norm = dinv[s] * w[e] * dinv[t];
        float4 v = *(const float4*)&h1[s * D_HID + q];
        float* o = &agg1[t * D_HID + q];
        __hip_atomic_fetch_add(o + 0, norm * v.x, __ATOMIC_RELAXED, __HIP_MEMORY_SCOPE_AGENT);
        __hip_atomic_fetch_add(o + 1, norm * v.y, __ATOMIC_RELAXED, __HIP_MEMORY_SCOPE_AGENT);
        __hip_atomic_fetch_add(o + 2, norm * v.z, __ATOMIC_RELAXED, __HIP_MEMORY_SCOPE_AGENT);
        __hip_atomic_fetch_add(o + 3, norm * v.w, __ATOMIC_RELAXED, __HIP_MEMORY_SCOPE_AGENT);
    }
}

// ---------------- epilogue 1: + self-loop + bias, leaky-relu (in place) -------
__global__ __launch_bounds__(256) void k_post1(float* __restrict__ agg1,
                                               const float* __restrict__ h1,
                                               const float* __restrict__ dinv,
                                               const float* __restrict__ b1) {
    int g = blockIdx.x * 256 + threadIdx.x;  // N*64 = 6.4M
    if (g < N_NODES * D_HID) {
        int i = g >> 6, d = g & 63;
        float di = dinv[i];
        float v = agg1[g] + di * di * h1[g] + b1[d];
        agg1[g] = (v > 0.0f) ? v : 0.01f * v;   // leaky_relu(0.01)
    }
}

// ---------------- layer-2 GEMM: h2 = h1f @ W2 (64 -> 4) ----------------
__global__ __launch_bounds__(256) void k_gemm2(const float* __restrict__ h1f,
                                               const float* __restrict__ W2,
                                               float* __restrict__ h2) {
    __shared__ float4 sW2[D_HID];
    if (threadIdx.x < D_HID)
        sW2[threadIdx.x] = *(const float4*)&W2[threadIdx.x * 4];
    __syncthreads();
    int i = blockIdx.x * 256 + threadIdx.x;
    if (i < N_NODES) {
        const float* r = &h1f[i * D_HID];
        float ax = 0.f, ay = 0.f, az = 0.f, aw = 0.f;
#pragma unroll
        for (int k = 0; k < D_HID; ++k) {
            float xv = r[k];
            float4 wk = sW2[k];
            ax += xv * wk.x; ay += xv * wk.y; az += xv * wk.z; aw += xv * wk.w;
        }
        *(float4*)&h2[i * 4] = make_float4(ax, ay, az, aw);
    }
}

// ---------------- layer-2 scatter: one thread per edge, 4 dims ----------------
__global__ __launch_bounds__(256) void k_scatter2(const int* __restrict__ src,
                                                  const int* __restrict__ dst,
                                                  const float* __restrict__ w,
                                                  const float* __restrict__ dinv,
                                                  const float* __restrict__ h2,
                                                  float* __restrict__ agg2) {
    int e = blockIdx.x * 256 + threadIdx.x;
    if (e < N_EDGES) {
        int s = src[e], t = dst[e];
        float norm = dinv[s] * w[e] * dinv[t];
        float4 v = *(const float4*)&h2[s * 4];
        float* o = &agg2[t * 4];
        __hip_atomic_fetch_add(o + 0, norm * v.x, __ATOMIC_RELAXED, __HIP_MEMORY_SCOPE_AGENT);
        __hip_atomic_fetch_add(o + 1, norm * v.y, __ATOMIC_RELAXED, __HIP_MEMORY_SCOPE_AGENT);
        __hip_atomic_fetch_add(o + 2, norm * v.z, __ATOMIC_RELAXED, __HIP_MEMORY_SCOPE_AGENT);
        __hip_atomic_fetch_add(o + 3, norm * v.w, __ATOMIC_RELAXED, __HIP_MEMORY_SCOPE_AGENT);
    }
}

// ---------------- epilogue 2: + self-loop + bias, softmax over 4 ----------------
__global__ __launch_bounds__(256) void k_soft(const float* __restrict__ agg2,
                                              const float* __restrict__ h2,
                                              const float* __restrict__ dinv,
                                              const float* __restrict__ b2,
                                              float* __restrict__ out) {
    int i = blockIdx.x * 256 + threadIdx.x;
    if (i < N_NODES) {
        float di = dinv[i], di2 = di * di;
        float4 a = *(const float4*)&agg2[i * 4];
        float4 h = *(const float4*)&h2[i * 4];
        float v0 = a.x + di2 * h.x + b2[0];
        float v1 = a.y + di2 * h.y + b2[1];
        float v2 = a.z + di2 * h.z + b2[2];
        float v3 = a.w + di2 * h.w + b2[3];
        float m = fmaxf(fmaxf(v0, v1), fmaxf(v2, v3));
        float e0 = expf(v0 - m), e1 = expf(v1 - m), e2 = expf(v2 - m), e3 = expf(v3 - m);
        float inv = 1.0f / (e0 + e1 + e2 + e3);
        *(float4*)&out[i * 4] = make_float4(e0 * inv, e1 * inv, e2 * inv, e3 * inv);
    }
}

// ---------------- host side ----------------
extern "C" void kernel_launch(void* const* d_in, const int* in_sizes, int n_in,
                              void* d_out, int out_size, void* d_ws, size_t ws_size,
                              hipStream_t stream) {
    const float* x   = (const float*)d_in[0];          // [N,128]
    const int*   ei  = (const int*)  d_in[1];          // [2,E]
    const float* w   = (const float*)d_in[2];          // [E]
    const float* W1  = (const float*)d_in[3];          // [128,64]
    const float* b1  = (const float*)d_in[4];          // [64]
    const float* W2  = (const float*)d_in[5];          // [64,4]
    const float* b2  = (const float*)d_in[6];          // [4]
    const int* src = ei;
    const int* dst = ei + N_EDGES;
    float* out = (float*)d_out;

    // Workspace layout (all offsets 128B-aligned); total = 54,800,000 bytes.
    char* ws = (char*)d_ws;
    float* dinv = (float*)(ws + 0);           //   400,000 B  (deg -> dinv in place)
    float* h1   = (float*)(ws + 400000);      // 25,600,000 B
    float* agg1 = (float*)(ws + 26000000);    // 25,600,000 B (becomes h1f in place)
    float* h2   = (float*)(ws + 51600000);    //  1,600,000 B
    float* agg2 = (float*)(ws + 53200000);    //  1,600,000 B
    (void)in_sizes; (void)n_in; (void)out_size; (void)ws_size;

    // Accumulators must be re-zeroed every call (graph replays reuse buffers).
    hipMemsetAsync(agg1, 0, (size_t)N_NODES * D_HID * sizeof(float), stream);
    hipMemsetAsync(agg2, 0, (size_t)N_NODES * D_OUT * sizeof(float), stream);

    k_init_deg<<<(N_NODES + 255) / 256, 256, 0, stream>>>(dinv);
    k_edge_deg<<<N_EDGES / 256, 256, 0, stream>>>(dst, w, dinv);
    k_dinv    <<<(N_NODES + 255) / 256, 256, 0, stream>>>(dinv);

    k_gemm1   <<<N_NODES / 32, 256, 0, stream>>>(x, W1, h1);          // 3125 blocks
    k_scatter1<<<(N_EDGES * 16) / 256, 256, 0, stream>>>(src, dst, w, dinv, h1, agg1);
    k_post1   <<<(N_NODES * D_HID) / 256, 256, 0, stream>>>(agg1, h1, dinv, b1);

    k_gemm2   <<<(N_NODES + 255) / 256, 256, 0, stream>>>(agg1, W2, h2);
    k_scatter2<<<N_EDGES / 256, 256, 0, stream>>>(src, dst, w, dinv, h2, agg2);
    k_soft    <<<(N_NODES + 255) / 256, 256, 0, stream>>>(agg2, h2, dinv, b2, out);
}